// nmNet_head_link_d1d2_2_12695923327495
// MI455X (gfx1250) — compile-verified
//
#include <hip/hip_runtime.h>

typedef __attribute__((ext_vector_type(2))) float v2f;
typedef __attribute__((ext_vector_type(8))) float v8f;

// Problem constants (from the reference)
#define BB   16
#define CIN  32
#define CC   64
#define HH   128
#define WW   128
#define KT   288          // CIN * 3 * 3, matches OIHW flattening of weights
#define NSTEPS 20

#define BSTRIDE 40        // floats per im2col row: 32 cols + 8 pad
                          // 2-row stride = 320B = 80 banks = +16 (mod 64)
                          // -> the two 16-lane groups hit disjoint bank sets

__global__ __launch_bounds__(256) void nmode_conv_rk4_fused(
    const float* __restrict__ xs,    // (16,32,128,128)
    const float* __restrict__ yd,    // (16,64,128,128)
    const float* __restrict__ wgt,   // (64,32,3,3) == (64,288) flat in K
    const float* __restrict__ bias,  // (64,)
    float* __restrict__ out)         // (16,64,128,128)
{
    __shared__ float lds_b[KT * BSTRIDE];        // 46080 B im2col tile

    const int blk = blockIdx.x;                  // 0..8191
    const int w32 = (blk & 3) * 32;              // 32-wide strip of W
    const int h   = (blk >> 2) & (HH - 1);
    const int b   = blk >> 9;                    // 512 blocks per batch image

    const int tid  = threadIdx.x;
    const int wave = tid >> 5;                   // 0..7
    const int lane = tid & 31;

    // Keep the (tiny, reused-by-everyone) weight table warm in cache.
    __builtin_prefetch(wgt + ((wave & 3) * 16 + (lane & 15)) * KT, 0, 3);

    // ---- stage im2col tile: lds_b[k][col] = x[b][cin][h-1+kh][w32-1+col+kw]
    // For fixed k, col 0..31 is contiguous in x -> coalesced 128B segments.
    for (int i = tid; i < KT * 32; i += 256) {
        const int k   = i >> 5;
        const int col = i & 31;
        const int cin = k / 9;
        const int r   = k - cin * 9;
        const int kh  = r / 3;
        const int kw  = r - kh * 3;
        const int gh  = h - 1 + kh;
        const int gw  = w32 - 1 + col + kw;
        float v = 0.0f;
        if (gh >= 0 && gh < HH && gw >= 0 && gw < WW)
            v = xs[((b * CIN + cin) * HH + gh) * WW + gw];
        lds_b[k * BSTRIDE + col] = v;
    }
    __syncthreads();

    // ---- per-wave 16(channels) x 16(pixels) tile -------------------------
    const int c0    = (wave & 3) * 16;           // channel tile base
    const int w0    = w32 + (wave >> 2) * 16;    // global w of pixel tile
    const int n     = lane & 15;                 // N position / A row (M)
    const int kslot = (lane >> 4) * 2;           // K sub-slot: 0 or 2

    // A: lane holds w[(c0+m)*288 + k0+kslot .. +1]; even offset -> aligned b64
    const float* aptr = wgt + (c0 + n) * KT + kslot;
    // B: lane holds lds_b[k0+kslot][lcol], lds_b[k0+kslot+1][lcol]
    const float* bptr = lds_b + kslot * BSTRIDE + (w0 - w32) + n;

    v8f acc = {};
#pragma unroll
    for (int k0 = 0; k0 < KT; k0 += 4) {
        const v2f a = *(const v2f*)(aptr + k0);          // global_load_b64 +imm
        v2f bm;
        bm.x = bptr[k0 * BSTRIDE];                       // ds_load_b32 +imm
        bm.y = bptr[(k0 + 1) * BSTRIDE];                 // ds_load_b32 +imm
        acc = __builtin_amdgcn_wmma_f32_16x16x4_f32(
            /*neg_a=*/false, a, /*neg_b=*/false, bm,
            /*c_mod=*/(short)0, acc, /*reuse_a=*/false, /*reuse_b=*/false);
    }

    // ---- bias + fused RK4 ODE on the tile held in registers --------------
    const int m_off = (lane >> 4) * 8;           // C/D layout: vgpr r -> M = r + m_off
    const float hs  = 1.0f / (float)NSTEPS;

    float g[8], y[8];
#pragma unroll
    for (int r = 0; r < 8; ++r) {
        const int c = c0 + r + m_off;
        g[r] = acc[r] + bias[c];
        y[r] = yd[((b * CC + c) * HH + h) * WW + w0 + n];
    }

    for (int s = 0; s < NSTEPS; ++s) {
#pragma unroll
        for (int r = 0; r < 8; ++r) {
            const float gg = g[r];
            const float yy = y[r];
            float t1 = __sinf(yy + gg);
            const float k1 = -yy + t1 * t1;
            const float y2 = yy + 0.5f * hs * k1;
            float t2 = __sinf(y2 + gg);
            const float k2 = -y2 + t2 * t2;
            const float y3 = yy + 0.5f * hs * k2;
            float t3 = __sinf(y3 + gg);
            const float k3 = -y3 + t3 * t3;
            const float y4 = yy + hs * k3;
            float t4 = __sinf(y4 + gg);
            const float k4 = -y4 + t4 * t4;
            y[r] = yy + (hs / 6.0f) * (k1 + 2.0f * k2 + 2.0f * k3 + k4);
        }
    }

#pragma unroll
    for (int r = 0; r < 8; ++r) {
        const int c = c0 + r + m_off;
        out[((b * CC + c) * HH + h) * WW + w0 + n] = y[r];
    }
}

extern "C" void kernel_launch(void* const* d_in, const int* in_sizes, int n_in,
                              void* d_out, int out_size, void* d_ws, size_t ws_size,
                              hipStream_t stream) {
    (void)in_sizes; (void)n_in; (void)d_ws; (void)ws_size; (void)out_size;
    const float* xs   = (const float*)d_in[0];   // x_shallow (16,32,128,128)
    const float* yd   = (const float*)d_in[1];   // y_deep    (16,64,128,128)
    const float* wgt  = (const float*)d_in[2];   // conv_w    (64,32,3,3)
    const float* bias = (const float*)d_in[3];   // conv_b    (64,)
    float* outp = (float*)d_out;

    const int grid = BB * HH * (WW / 32);        // 8192 blocks of 256 threads
    nmode_conv_rk4_fused<<<grid, 256, 0, stream>>>(xs, yd, wgt, bias, outp);
}